// GNNq_2027224563935
// MI455X (gfx1250) — compile-verified
//
#include <hip/hip_runtime.h>

// ---------------------------------------------------------------------------
// 2-layer GCN for gfx1250 (MI455X, wave32).
//   W1t = transpose(W1)               (once per call, 64K elems)
//   xw  = x @ W1     fp32 WMMA (v_wmma_f32_16x16x4_f32). W1t K-chunks are
//                    DMA'd into LDS by the Tensor Data Mover with LDS padding
//                    (bank-conflict-free, tensor_load_to_lds + s_wait_tensorcnt);
//                    B fragments are single aligned ds_load_b64.
//   h   = scatter_add(ev * xw[col])   edge-parallel, global_atomic_add_f32
//   hw  = relu(h) @ W2                fp32 WMMA, ReLU fused into A-load
//   out = scatter_add(ev * hw[col])
// ---------------------------------------------------------------------------

typedef __attribute__((ext_vector_type(2))) float        v2f;
typedef __attribute__((ext_vector_type(8))) float        v8f;
typedef __attribute__((ext_vector_type(4))) unsigned int v4u;
typedef __attribute__((ext_vector_type(4))) int          v4i;
typedef __attribute__((ext_vector_type(8))) int          v8i;

#define NF 512       // input features
#define NH 128       // hidden dim
#define NC 16        // classes
#define KC 64        // K-chunk staged in LDS per TDM transfer
#define LDSPITCH 68  // KC + 4 pad DWORDs per row (TDM pad) -> conflict-free

// ------------------------------- zero fill ---------------------------------
__global__ void zero_f4(float4* __restrict__ p, long n4) {
    long i      = (long)blockIdx.x * blockDim.x + threadIdx.x;
    long stride = (long)gridDim.x * blockDim.x;
    float4 z;  z.x = z.y = z.z = z.w = 0.0f;
    for (; i < n4; i += stride) p[i] = z;
}

// --------------------- W1t[n][k] = W1[k][n]  (write-coalesced) -------------
__global__ void transpose_w1(const float* __restrict__ W, float* __restrict__ Wt) {
    int idx = blockIdx.x * blockDim.x + threadIdx.x;   // over NH*NF
    if (idx >= NH * NF) return;
    int n = idx / NF;
    int k = idx - n * NF;
    Wt[idx] = W[k * NH + n];
}

// ---------------- TDM: DMA a 2D fp32 tile (rows x cols) into LDS -----------
// D# per CDNA5 ISA ch.8. group0 = {count=1, lds_addr, global_addr, type=2}.
// group1 = {data_size=4B, pad_enable, pad_interval=64 DW, pad_amount=4 DW,
//           tensor_dim0/1, tile_dim0=cols, tile_dim1=rows, dim0_stride}.
// Padding inserts 4 DWORDs after each 64-DWORD row -> LDS pitch 68 floats.
__device__ __forceinline__ void tdm_load_2d_pad(unsigned lds_off, const void* gptr,
                                                unsigned rows, unsigned cols,
                                                unsigned stride_elems,
                                                unsigned tensor_cols) {
    unsigned long long ga = (unsigned long long)(uintptr_t)gptr;
    v4u g0;
    g0[0] = 1u;                                        // count=1 (valid D#)
    g0[1] = lds_off;                                   // LDS byte address
    g0[2] = (unsigned)(ga & 0xFFFFFFFFu);              // global_addr[31:0]
    g0[3] = (unsigned)((ga >> 32) & 0x01FFFFFFu) | (2u << 30);  // type=2
    v8i g1;
    g1[0] = (int)((2u << 16) |                         // data_size=2 -> 4 bytes
                  (1u << 20) |                         // pad_enable
                  (5u << 22) |                         // pad_interval: 64 DWORDs
                  (3u << 25));                         // pad_amount: 4 DWORDs
    g1[1] = (int)((tensor_cols & 0xFFFFu) << 16);      // tensor_dim0[15:0]
    g1[2] = (int)(((tensor_cols >> 16) & 0xFFFFu) |
                  ((rows & 0xFFFFu) << 16));           // dim0 hi | tensor_dim1 lo
    g1[3] = (int)(((rows >> 16) & 0xFFFFu) |
                  ((cols & 0xFFFFu) << 16));           // dim1 hi | tile_dim0
    g1[4] = (int)(rows & 0xFFFFu);                     // tile_dim1 (tile_dim2=0)
    g1[5] = (int)stride_elems;                         // tensor_dim0_stride lo32
    g1[6] = 0;
    g1[7] = 0;
    v4i gz = {0, 0, 0, 0};
#if defined(__clang_major__) && (__clang_major__ >= 23)
    v8i gz8 = {0, 0, 0, 0, 0, 0, 0, 0};
    __builtin_amdgcn_tensor_load_to_lds(g0, g1, gz, gz, gz8, 0);
#else
    __builtin_amdgcn_tensor_load_to_lds(g0, g1, gz, gz, 0);
#endif
}

// ------------------------ GEMM1: Y[N,128] = X[N,512] @ W[512,128] ----------
// 8 waves/block; each wave owns a 16-row x 128-col strip (8 WMMA accums).
// Wt (column chunks of W1t) staged into LDS by wave 0 via TDM; B fragment
// (K,K+1 pair) is one aligned ds_load_b64; A streamed once from global.
__global__ __launch_bounds__(256) void gemm1_wmma(const float* __restrict__ X,
                                                  const float* __restrict__ Wt,
                                                  float* __restrict__ Y,
                                                  int nNodes) {
    __shared__ float bs[NH * LDSPITCH];                // 34 KB padded B chunk
    const int wave = threadIdx.x >> 5;
    const int lane = threadIdx.x & 31;
    const int mTiles = nNodes >> 4;
    int mt = blockIdx.x * 8 + wave;
    const bool valid = (mt < mTiles);
    if (!valid) mt = mTiles - 1;    // clamp: must keep running (barriers below)
    const int m0    = mt << 4;
    const int l15   = lane & 15;          // M-row (A) / N-col (B,C)
    const int khalf = (lane >> 4) << 1;   // lanes 16-31 hold K pair {2,3}

    v8f acc[8];
    const v8f vzero = {0.f,0.f,0.f,0.f,0.f,0.f,0.f,0.f};
#pragma unroll
    for (int i = 0; i < 8; ++i) acc[i] = vzero;

    const float* arow = X + (long)(m0 + l15) * NF + khalf;
    const unsigned lds_off = (unsigned)(uintptr_t)&bs[0];
    // per-lane LDS base: row n=l15 (nt adds 16*LDSPITCH), column khalf
    const float* bbase = bs + l15 * LDSPITCH + khalf;

    for (int chunk = 0; chunk < NF / KC; ++chunk) {
        if (wave == 0) {                               // one TDM per block/chunk
            tdm_load_2d_pad(lds_off, Wt + chunk * KC,
                            NH, KC, NF, NF - chunk * KC);
            __builtin_amdgcn_s_wait_tensorcnt(0);
        }
        __syncthreads();                               // publish LDS chunk

        const int kbase = chunk * KC;
#pragma unroll 4
        for (int kk = 0; kk < KC; kk += 4) {
            v2f a;                                     // A frag: 16x4 fp32
            a.x = arow[kbase + kk];
            a.y = arow[kbase + kk + 1];
#pragma unroll
            for (int nt = 0; nt < 8; ++nt) {           // B frag: one ds_load_b64
                v2f b = *(const v2f*)(bbase + nt * 16 * LDSPITCH + kk);
                acc[nt] = __builtin_amdgcn_wmma_f32_16x16x4_f32(
                    false, a, false, b, (short)0, acc[nt], false, false);
            }
        }
        __syncthreads();                               // drain before next DMA
    }

    if (valid) {
        // C/D layout: VGPR r -> M=r (lanes 0-15) / M=r+8 (lanes 16-31), N=l15
        const int rofs = (lane >> 4) << 3;
#pragma unroll
        for (int nt = 0; nt < 8; ++nt)
#pragma unroll
            for (int r = 0; r < 8; ++r)
                Y[(long)(m0 + rofs + r) * NH + nt * 16 + l15] = acc[nt][r];
    }
}

// --------------- GEMM2: Y[N,16] = relu(H[N,128]) @ W[128,16] ---------------
__global__ __launch_bounds__(256) void gemm2_wmma(const float* __restrict__ H,
                                                  const float* __restrict__ W,
                                                  float* __restrict__ Y,
                                                  int nNodes) {
    const int wave = threadIdx.x >> 5;
    const int lane = threadIdx.x & 31;
    const int mTiles = nNodes >> 4;
    const int mt = blockIdx.x * 8 + wave;
    if (mt >= mTiles) return;                          // no barriers here: safe
    const int m0    = mt << 4;
    const int l15   = lane & 15;
    const int khalf = (lane >> 4) << 1;

    v8f acc = {0.f,0.f,0.f,0.f,0.f,0.f,0.f,0.f};
    const float* arow = H + (long)(m0 + l15) * NH + khalf;

    for (int k0 = 0; k0 < NH; k0 += 4) {
        v2f a;                               // ReLU fused into A-fragment load
        a.x = fmaxf(arow[k0],     0.0f);
        a.y = fmaxf(arow[k0 + 1], 0.0f);
        const float* bp = W + (long)(k0 + khalf) * NC + l15;
        v2f b;
        b.x = bp[0];
        b.y = bp[NC];
        acc = __builtin_amdgcn_wmma_f32_16x16x4_f32(
            false, a, false, b, (short)0, acc, false, false);
    }

    const int rofs = (lane >> 4) << 3;
#pragma unroll
    for (int r = 0; r < 8; ++r)
        Y[(long)(m0 + rofs + r) * NC + l15] = acc[r];
}

// -------- SpMM (D=128): one wave per edge, float4 gather, atomic scatter ---
__global__ void spmm_scatter128(const float* __restrict__ feat,
                                const float* __restrict__ ev,
                                const int* __restrict__ er,
                                const int* __restrict__ ec,
                                float* __restrict__ acc, int nE) {
    const long gid = (long)blockIdx.x * blockDim.x + threadIdx.x;
    const int  e    = (int)(gid >> 5);
    const int  lane = (int)(gid & 31);
    if (e >= nE) return;
    const int   row = er[e];
    const int   col = ec[e];
    const float v   = ev[e];
    const float4* src = (const float4*)(feat + (long)col * NH);
    float4 m = src[lane];                       // 32 lanes x float4 = 128 f32
    float* dst = acc + (long)row * NH + lane * 4;
    atomicAdd(dst + 0, v * m.x);
    atomicAdd(dst + 1, v * m.y);
    atomicAdd(dst + 2, v * m.z);
    atomicAdd(dst + 3, v * m.w);
}

// -------- SpMM (D=16): 16 lanes per edge (2 edges per wave) ----------------
__global__ void spmm_scatter16(const float* __restrict__ feat,
                               const float* __restrict__ ev,
                               const int* __restrict__ er,
                               const int* __restrict__ ec,
                               float* __restrict__ out, int nE) {
    const long gid = (long)blockIdx.x * blockDim.x + threadIdx.x;
    const int  e   = (int)(gid >> 4);
    const int  f   = (int)(gid & 15);
    if (e >= nE) return;
    const int   row = er[e];
    const int   col = ec[e];
    const float v   = ev[e];
    atomicAdd(out + (long)row * NC + f, v * feat[(long)col * NC + f]);
}

// ---------------------------------------------------------------------------
extern "C" void kernel_launch(void* const* d_in, const int* in_sizes, int n_in,
                              void* d_out, int out_size, void* d_ws, size_t ws_size,
                              hipStream_t stream) {
    const float* x  = (const float*)d_in[0];
    const float* W1 = (const float*)d_in[1];
    const float* W2 = (const float*)d_in[2];
    const float* ev = (const float*)d_in[3];
    const int*   er = (const int*)d_in[4];
    const int*   ec = (const int*)d_in[5];

    const int nNodes = in_sizes[0] / NF;   // 100000
    const int nE     = in_sizes[3];        // 1600000
    float* out = (float*)d_out;

    // Workspace: [ xw : N*NH | hacc : N*NH | W1t : NF*NH ].  hw aliases xw
    // (xw is dead once SpMM1 has consumed it).
    float* xw   = (float*)d_ws;
    float* hacc = xw   + (size_t)nNodes * NH;
    float* w1t  = hacc + (size_t)nNodes * NH;
    float* hw   = xw;

    const int mTiles = nNodes >> 4;
    const int gGemm  = (mTiles + 7) / 8;

    // 1) zero the atomic accumulators (every call: graph replays re-accumulate)
    //    and build W1t (independent kernels, same stream).
    {
        long n4h = (long)nNodes * NH / 4;
        long n4o = (long)nNodes * NC / 4;
        zero_f4<<<2048, 256, 0, stream>>>((float4*)hacc, n4h);
        zero_f4<<<1024, 256, 0, stream>>>((float4*)out,  n4o);
        transpose_w1<<<(NH * NF + 255) / 256, 256, 0, stream>>>(W1, w1t);
    }

    // 2) xw = x @ W1   (fp32 WMMA, TDM-staged padded B in LDS)
    gemm1_wmma<<<gGemm, 256, 0, stream>>>(x, w1t, xw, nNodes);

    // 3) hacc += ev * xw[col]  (scatter)
    {
        long threads = (long)nE * 32;
        int  grid    = (int)((threads + 255) / 256);
        spmm_scatter128<<<grid, 256, 0, stream>>>(xw, ev, er, ec, hacc, nE);
    }

    // 4) hw = relu(hacc) @ W2   (fp32 WMMA, ReLU fused)
    gemm2_wmma<<<gGemm, 256, 0, stream>>>(hacc, W2, hw, nNodes);

    // 5) out += ev * hw[col]  (scatter)
    {
        long threads = (long)nE * 16;
        int  grid    = (int)((threads + 255) / 256);
        spmm_scatter16<<<grid, 256, 0, stream>>>(hw, ev, er, ec, out, nE);
    }
}